// Attention_4114578669666
// MI455X (gfx1250) — compile-verified
//
#include <hip/hip_runtime.h>
#include <hip/hip_bf16.h>
#include <math.h>

typedef __attribute__((ext_vector_type(2))) float v2f;
typedef __attribute__((ext_vector_type(8))) float v8f;

#define B 32
#define D_MODEL 4096
#define H 32
#define HKV 8
#define HD 128
#define KVLEN 2048
#define BLOCK 256
#define NBPS (KVLEN / BLOCK)
#define EPS 1e-6f
#define THETA 10000.0f
#define AWAVES 16            // waves per attention block
#define KPW (KVLEN / AWAVES) // 128 keys per wave

// ---------------------------------------------------------------------------
// GEMM: Y[32 x N] = X[32 x K] @ W[K x N], f32 WMMA 16x16x4.
// One wave computes a 32x16 output tile (two 16x16 accumulators) so each
// 16-column weight slab is streamed from HBM exactly once.
// ---------------------------------------------------------------------------
__global__ void gemm32_wmma(const float* __restrict__ X,
                            const float* __restrict__ W,
                            float* __restrict__ Y, int N, int K) {
  const int lane = threadIdx.x & 31;
  const int wv   = threadIdx.x >> 5;
  const int tile = blockIdx.x * (blockDim.x >> 5) + wv;
  const int n0   = tile * 16;
  if (n0 >= N) return;

  const int m  = lane & 15;
  const int kh = lane >> 4;   // 0: K pair {0,1}, 1: K pair {2,3}

  v8f c0 = {};
  v8f c1 = {};

  for (int kb = 0; kb < K; kb += 4) {
    const int kk = kb + 2 * kh;
    v2f a0 = *(const v2f*)(X + (size_t)m * K + kk);
    v2f a1 = *(const v2f*)(X + (size_t)(m + 16) * K + kk);
    v2f bf;
    bf.x = W[(size_t)kk * N + n0 + m];
    bf.y = W[(size_t)(kk + 1) * N + n0 + m];
    c0 = __builtin_amdgcn_wmma_f32_16x16x4_f32(false, a0, false, bf,
                                               (short)0, c0, false, false);
    c1 = __builtin_amdgcn_wmma_f32_16x16x4_f32(false, a1, false, bf,
                                               (short)0, c1, false, false);
  }

  const int col   = n0 + m;
  const int rbase = kh * 8;
#pragma unroll
  for (int r = 0; r < 8; ++r) {
    Y[(size_t)(rbase + r) * N + col]      = c0[r];
    Y[(size_t)(rbase + r + 16) * N + col] = c1[r];
  }
}

// ---------------------------------------------------------------------------
// RMSNorm + RoPE for Q (32 heads) and K (8 heads), in-place in workspace.
// ---------------------------------------------------------------------------
__global__ void norm_rope(float* __restrict__ qws, float* __restrict__ kws,
                          const float* __restrict__ qn_w,
                          const float* __restrict__ kn_w,
                          const int* __restrict__ input_pos) {
  const int b = blockIdx.x;
  const int t = threadIdx.x;  // 0..127
  __shared__ float red[HD];
  __shared__ float xbuf[HD];

  const float pos = (float)input_pos[b];
  float cang = 0.f, sang = 0.f;
  if (t < HD / 2) {
    float inv = __powf(THETA, -(float)t / (float)(HD / 2));
    float ang = pos * inv;
    cang = __cosf(ang);
    sang = __sinf(ang);
  }

  for (int h = 0; h < H + HKV; ++h) {
    float* x;
    float w;
    if (h < H) {
      x = qws + ((size_t)b * H + h) * HD;
      w = qn_w[t];
    } else {
      x = kws + ((size_t)b * HKV + (h - H)) * HD;
      w = kn_w[t];
    }
    float v = x[t];
    red[t] = v * v;
    __syncthreads();
    for (int s = HD / 2; s > 0; s >>= 1) {
      if (t < s) red[t] += red[t + s];
      __syncthreads();
    }
    float scale = rsqrtf(red[0] / (float)HD + EPS);
    xbuf[t] = v * scale * w;
    __syncthreads();
    if (t < HD / 2) {
      float x1 = xbuf[t];
      float x2 = xbuf[t + HD / 2];
      x[t]          = x1 * cang - x2 * sang;
      x[t + HD / 2] = x2 * cang + x1 * sang;
    }
    __syncthreads();
  }
}

// ---------------------------------------------------------------------------
// WMMA flash-decode attention. Grid (B, HKV), 16 waves/block; wave w owns
// keys [w*128, w*128+128) in 16-key tiles:
//   scores: S(16x16) = Qpad(16x128) @ K^T  via 32x v_wmma_f32_16x16x4_f32
//   online softmax in the C layout (VGPR g = row, lane = key)
//   P transposed through per-wave LDS tile, then
//   O(16x128) += P(16x16) @ V(16x128)      via 32x WMMA into 8 accumulators.
// The freshly written K/V token is substituted when slot == slot_mapping[b]
// (input caches are never mutated).
// ---------------------------------------------------------------------------
__global__ __launch_bounds__(32 * AWAVES)
void attn_decode_wmma(const float* __restrict__ qws,
                      const float* __restrict__ kws,
                      const float* __restrict__ vws,
                      const float* __restrict__ kcache,
                      const float* __restrict__ vcache,
                      const int* __restrict__ block_tables,
                      const int* __restrict__ slot_mapping,
                      const int* __restrict__ context_lens,
                      float* __restrict__ outws) {
  const int b    = blockIdx.x;
  const int kvh  = blockIdx.y;
  const int tid  = threadIdx.x;
  const int lane = tid & 31;
  const int w    = tid >> 5;
  const int ctx  = context_lens[b];
  const int myslot = slot_mapping[b];
  const float scale = 0.08838834764831845f;  // 1/sqrt(128)

  __shared__ float lds_q[16][HD];                // rows 0-3 scaled Q, 4-15 zero
  __shared__ float lds_p[AWAVES][16][16];        // [wave][row g][key]
  __shared__ float sacc[AWAVES][4][HD];
  __shared__ float sm[AWAVES][4];
  __shared__ float sl[AWAVES][4];

  // Stage padded, pre-scaled Q tile (shared by all waves).
  for (int i = tid; i < 16 * HD; i += 32 * AWAVES) {
    const int r = i >> 7, d = i & (HD - 1);
    float v = 0.f;
    if (r < 4) v = qws[(((size_t)b * H) + kvh * 4 + r) * HD + d] * scale;
    lds_q[r][d] = v;
  }
  // Zero P rows 4-15 of this wave's tile (written once, never touched again).
  for (int i = lane; i < 12 * 16; i += 32) {
    lds_p[w][4 + (i >> 4)][i & 15] = 0.f;
  }
  __syncthreads();

  const int m16 = lane & 15;
  const int kh  = lane >> 4;

  float m_r[4] = {-1e30f, -1e30f, -1e30f, -1e30f};
  float ls[4]  = {0.f, 0.f, 0.f, 0.f};
  v8f acc[8];
#pragma unroll
  for (int t = 0; t < 8; ++t) { v8f z = {}; acc[t] = z; }

  const float* ksub = kws + (((size_t)b * HKV) + kvh) * HD;  // current token
  const float* vsub = vws + (((size_t)b * HKV) + kvh) * HD;

  const int kbase = w * KPW;
  for (int t = 0; t < KPW / 16; ++t) {
    const int l0 = kbase + t * 16;
    const int bt = block_tables[b * NBPS + (l0 >> 8)];       // tile-uniform
    const int slotbase = bt * BLOCK + (l0 & (BLOCK - 1));

    // ---- scores: S = Qpad @ K^T ------------------------------------------
    const int kslot = slotbase + m16;
    const float* krow = (kslot == myslot)
                            ? ksub
                            : kcache + ((size_t)kslot * HKV + kvh) * HD;
    v8f cs = {};
#pragma unroll 8
    for (int kb = 0; kb < HD; kb += 4) {
      const int kk = kb + 2 * kh;
      v2f a  = *(const v2f*)&lds_q[m16][kk];
      v2f bf = *(const v2f*)(krow + kk);
      cs = __builtin_amdgcn_wmma_f32_16x16x4_f32(false, a, false, bf,
                                                 (short)0, cs, false, false);
    }

    // ---- online softmax (lane = key, VGPR g = row) -----------------------
    const bool valid = (l0 + m16) < ctx;
#pragma unroll
    for (int g = 0; g < 4; ++g) {
      float s = valid ? cs[g] : -1e30f;
      float tm = s;
#pragma unroll
      for (int off = 1; off <= 8; off <<= 1)
        tm = fmaxf(tm, __shfl_xor(tm, off, 32));
      tm = __shfl(tm, 0, 32);                 // wave-uniform tile max
      const float nm    = fmaxf(m_r[g], tm);
      const float alpha = __expf(m_r[g] - nm);
      m_r[g] = nm;
      float p = __expf(s - nm);
      p = (valid && lane < 16) ? p : 0.f;
      float ps = p;
#pragma unroll
      for (int off = 16; off >= 1; off >>= 1)
        ps += __shfl_xor(ps, off, 32);
      ls[g] = ls[g] * alpha + ps;
      if (lane < 16) lds_p[w][g][lane] = p;
#pragma unroll
      for (int tt = 0; tt < 8; ++tt) acc[tt][g] *= alpha;
    }
    __builtin_amdgcn_wave_barrier();  // keep LDS p stores before frag reads

    // ---- O += P @ V -------------------------------------------------------
#pragma unroll
    for (int ks = 0; ks < 4; ++ks) {
      const int koff = 4 * ks + 2 * kh;
      v2f a = *(const v2f*)&lds_p[w][m16][koff];
      const int s0 = slotbase + koff;
      const int s1 = s0 + 1;
      const float* vr0 = (s0 == myslot)
                             ? vsub
                             : vcache + ((size_t)s0 * HKV + kvh) * HD;
      const float* vr1 = (s1 == myslot)
                             ? vsub
                             : vcache + ((size_t)s1 * HKV + kvh) * HD;
#pragma unroll
      for (int tt = 0; tt < 8; ++tt) {
        v2f bf;
        bf.x = vr0[tt * 16 + m16];
        bf.y = vr1[tt * 16 + m16];
        acc[tt] = __builtin_amdgcn_wmma_f32_16x16x4_f32(false, a, false, bf,
                                                        (short)0, acc[tt],
                                                        false, false);
      }
    }
  }

  // ---- cross-wave combine -------------------------------------------------
#pragma unroll
  for (int g = 0; g < 4; ++g) {
    if (lane == 0) { sm[w][g] = m_r[g]; sl[w][g] = ls[g]; }
    if (lane < 16) {
#pragma unroll
      for (int tt = 0; tt < 8; ++tt) sacc[w][g][tt * 16 + lane] = acc[tt][g];
    }
  }
  __syncthreads();

  if (tid < HD) {
    for (int g = 0; g < 4; ++g) {
      float M = -1e30f;
      for (int ww = 0; ww < AWAVES; ++ww) M = fmaxf(M, sm[ww][g]);
      float L = 0.f, o = 0.f;
      for (int ww = 0; ww < AWAVES; ++ww) {
        const float e = __expf(sm[ww][g] - M);
        L += e * sl[ww][g];
        o += e * sacc[ww][g][tid];
      }
      outws[(((size_t)b * H) + kvh * 4 + g) * HD + tid] = o / L;
    }
  }
}

// ---------------------------------------------------------------------------
extern "C" void kernel_launch(void* const* d_in, const int* in_sizes, int n_in,
                              void* d_out, int out_size, void* d_ws, size_t ws_size,
                              hipStream_t stream) {
  const float* seqs      = (const float*)d_in[0];
  const float* Wq        = (const float*)d_in[1];
  const float* Wk        = (const float*)d_in[2];
  const float* Wv        = (const float*)d_in[3];
  const float* Wo        = (const float*)d_in[4];
  const float* qn_w      = (const float*)d_in[5];
  const float* kn_w      = (const float*)d_in[6];
  const float* k_cache   = (const float*)d_in[7];
  const float* v_cache   = (const float*)d_in[8];
  const int*   input_pos = (const int*)d_in[9];
  const int*   slot_map  = (const int*)d_in[10];
  const int*   block_tbl = (const int*)d_in[11];
  const int*   ctx_lens  = (const int*)d_in[12];
  float* out = (float*)d_out;

  float* ws  = (float*)d_ws;
  float* qws = ws;                         // B * H*HD
  float* kws = qws + (size_t)B * H * HD;   // B * HKV*HD
  float* vws = kws + (size_t)B * HKV * HD; // B * HKV*HD
  float* aws = vws + (size_t)B * HKV * HD; // B * H*HD

  gemm32_wmma<<<64, 128, 0, stream>>>(seqs, Wq, qws, H * HD, D_MODEL);
  gemm32_wmma<<<16, 128, 0, stream>>>(seqs, Wk, kws, HKV * HD, D_MODEL);
  gemm32_wmma<<<16, 128, 0, stream>>>(seqs, Wv, vws, HKV * HD, D_MODEL);

  norm_rope<<<B, HD, 0, stream>>>(qws, kws, qn_w, kn_w, input_pos);

  attn_decode_wmma<<<dim3(B, HKV), 32 * AWAVES, 0, stream>>>(
      qws, kws, vws, k_cache, v_cache, block_tbl, slot_map, ctx_lens, aws);

  gemm32_wmma<<<64, 128, 0, stream>>>(aws, Wo, out, D_MODEL, H * HD);
}